// RoutingCapsule_17308718203516
// MI455X (gfx1250) — compile-verified
//
#include <hip/hip_runtime.h>
#include <math.h>

typedef __attribute__((ext_vector_type(2))) float v2f;
typedef __attribute__((ext_vector_type(8))) float v8f;

#define B_BATCH  64
#define IN_CAPS  2048
#define IN_DIM   16
#define OUT_CAPS 32
#define OUT_DIM  16
#define I_CHUNK  32   // input capsules per workgroup

// Fused routing pass.
// grid = (IN_CAPS / I_CHUNK, B_BATCH/16), block = 256 (8 waves, wave32).
// Wave w owns output capsules [4w, 4w+4). For each input capsule i in the
// chunk it recomputes u_hat tiles with V_WMMA_F32_16X16X4_F32, then
//  ITER==0: s += (1/32) * u_hat
//  ITER>0 : a = sum_d u_hat * v  (16-lane shuffle reduce)
//           b_ij (+)= a ; c = softmax_o(b_ij) ; s += c * u_hat
// Per-workgroup s partials accumulate in registers, one atomic flush at end.
template <int ITER>
__global__ __launch_bounds__(256) void caps_pass(
    const float* __restrict__ x,   // (B, IN_CAPS, IN_DIM)
    const float* __restrict__ w,   // (IN_CAPS, OUT_CAPS, OUT_DIM, IN_DIM)
    const float* __restrict__ vin, // (B, OUT_CAPS, OUT_DIM) from prev squash
    float* s,                      // (B, OUT_CAPS, OUT_DIM) accum (pre-zeroed)
    float* bws)                    // (B, IN_CAPS, OUT_CAPS) routing logits
{
  __shared__ float a_lds[16][OUT_CAPS];
  __shared__ float c_lds[16][OUT_CAPS];

  const int tid   = threadIdx.x;
  const int wave  = tid >> 5;
  const int lane  = tid & 31;
  const int m     = lane & 15;   // d index / N column / M row
  const int hi    = lane >> 4;   // half-wave select
  const int i0    = blockIdx.x * I_CHUNK;
  const int b0    = blockIdx.y * 16;
  const int obase = wave * 4;

  v8f sacc[4];
#pragma unroll
  for (int ol = 0; ol < 4; ++ol) sacc[ol] = v8f{};

  // v[b, o, d] is i-invariant: hoist per-(tile-row, o) values out of the loop.
  float vv[4][8];
  if (ITER > 0) {
#pragma unroll
    for (int ol = 0; ol < 4; ++ol)
#pragma unroll
      for (int j = 0; j < 8; ++j)
        vv[ol][j] = vin[((size_t)(b0 + j + 8 * hi) * OUT_CAPS + (obase + ol)) * OUT_DIM + m];
  }

  for (int ii = 0; ii < I_CHUNK; ++ii) {
    const int i = i0 + ii;

    // A tile: X_i[b, k], 16x16 split into four 16x4 K-slices.
    // A layout (32-bit 16x4): VGPR0 = K=k0 (lanes 0-15) / K=k0+2 (lanes 16-31),
    //                         VGPR1 = K=k0+1 / K=k0+3.
    v2f A[4];
    {
      const float* xp = x + ((size_t)(b0 + m) * IN_CAPS + i) * IN_DIM;
#pragma unroll
      for (int t = 0; t < 4; ++t) {
        const int k = 4 * t + 2 * hi;
        A[t].x = xp[k];
        A[t].y = xp[k + 1];
      }
    }

    // u_hat tiles: one per owned output capsule.
    v8f u[4];
#pragma unroll
    for (int ol = 0; ol < 4; ++ol) {
      const int o = obase + ol;
      // B tile (4x16, K x N): B(k, n=d) = W[i, o, d, k]
      const float* wp = w + (((size_t)i * OUT_CAPS + o) * OUT_DIM + m) * IN_DIM;
      v8f acc = v8f{};
#pragma unroll
      for (int t = 0; t < 4; ++t) {
        const int k = 4 * t + 2 * hi;
        v2f Bm;
        Bm.x = wp[k];
        Bm.y = wp[k + 1];
        acc = __builtin_amdgcn_wmma_f32_16x16x4_f32(
            false, A[t], false, Bm, (short)0, acc, false, false);
      }
      u[ol] = acc;
    }

    if (ITER > 0) {
      // Agreement: a[b, o] = sum_d u_hat[b,i,o,d] * v[b,o,d]
#pragma unroll
      for (int ol = 0; ol < 4; ++ol) {
#pragma unroll
        for (int j = 0; j < 8; ++j) {
          float p = u[ol][j] * vv[ol][j];
          p += __shfl_xor(p, 1, 16);
          p += __shfl_xor(p, 2, 16);
          p += __shfl_xor(p, 4, 16);
          p += __shfl_xor(p, 8, 16);
          if (m == 0) a_lds[j + 8 * hi][obase + ol] = p;
        }
      }
      __syncthreads();

      // Logit update + softmax over the 32 output capsules (one row leader
      // per batch row; 32-wide serial pass, fully register-resident).
      if (tid < 16) {
        const int row = tid;
        const size_t bi = ((size_t)(b0 + row) * IN_CAPS + i) * OUT_CAPS;
        float t[OUT_CAPS];
        float mx = -1e30f;
#pragma unroll
        for (int o = 0; o < OUT_CAPS; ++o) {
          float val = a_lds[row][o];
          if (ITER == 2) val += bws[bi + o];  // b_ij = a0 + a1
          if (ITER == 1) bws[bi + o] = val;   // b_ij = a0 (fresh write)
          t[o] = val;
          mx = fmaxf(mx, val);
        }
        float sum = 0.0f;
#pragma unroll
        for (int o = 0; o < OUT_CAPS; ++o) {
          const float e = __expf(t[o] - mx);
          t[o] = e;
          sum += e;
        }
        const float inv = 1.0f / sum;
#pragma unroll
        for (int o = 0; o < OUT_CAPS; ++o) c_lds[row][o] = t[o] * inv;
      }
      __syncthreads();

      // s partial: s[b,o,d] += c[b,i,o] * u_hat[b,i,o,d]
#pragma unroll
      for (int ol = 0; ol < 4; ++ol)
#pragma unroll
        for (int j = 0; j < 8; ++j)
          sacc[ol][j] += c_lds[j + 8 * hi][obase + ol] * u[ol][j];
      __syncthreads();  // a_lds/c_lds reused next i
    } else {
      // iter 0: softmax(0) is uniform 1/OUT_CAPS
      constexpr float cu = 1.0f / OUT_CAPS;
#pragma unroll
      for (int ol = 0; ol < 4; ++ol)
#pragma unroll
        for (int j = 0; j < 8; ++j) sacc[ol][j] += cu * u[ol][j];
    }
  }

  // Flush chunk-local s partials (cross-workgroup reduction over i-chunks).
#pragma unroll
  for (int ol = 0; ol < 4; ++ol)
#pragma unroll
    for (int j = 0; j < 8; ++j)
      atomicAdd(&s[((size_t)(b0 + j + 8 * hi) * OUT_CAPS + (obase + ol)) * OUT_DIM + m],
                sacc[ol][j]);
}

__global__ void zero_kernel(float* p, int n) {
  const int i = blockIdx.x * blockDim.x + threadIdx.x;
  if (i < n) p[i] = 0.0f;
}

// v = s^3 / ((1+s^2) * sqrt(s^2 + 1e-8)); optionally re-zero s for next pass.
__global__ void squash_kernel(float* s, float* __restrict__ v, int clear_s, int n) {
  const int i = blockIdx.x * blockDim.x + threadIdx.x;
  if (i < n) {
    const float sv = s[i];
    const float sq = sv * sv;
    v[i] = sq * sv / ((1.0f + sq) * sqrtf(sq + 1e-8f));
    if (clear_s) s[i] = 0.0f;
  }
}

extern "C" void kernel_launch(void* const* d_in, const int* in_sizes, int n_in,
                              void* d_out, int out_size, void* d_ws, size_t ws_size,
                              hipStream_t stream) {
  (void)in_sizes; (void)n_in; (void)out_size; (void)ws_size;
  const float* x = (const float*)d_in[0];   // (64, 2048, 16)
  const float* w = (const float*)d_in[1];   // (2048, 32, 16, 16)
  float* out = (float*)d_out;               // (64, 32, 16, 1) flat

  char* ws = (char*)d_ws;
  float* bws = (float*)ws;                                       // 16.8 MB logits
  float* s   = (float*)(ws + (size_t)B_BATCH * IN_CAPS * OUT_CAPS * sizeof(float));
  float* v   = s + (size_t)B_BATCH * OUT_CAPS * OUT_DIM;

  const int NS = B_BATCH * OUT_CAPS * OUT_DIM;  // 32768
  const dim3 grid(IN_CAPS / I_CHUNK, B_BATCH / 16);
  const dim3 blk(256);
  const dim3 sgrid((NS + 255) / 256), sblk(256);

  zero_kernel<<<sgrid, sblk, 0, stream>>>(s, NS);

  caps_pass<0><<<grid, blk, 0, stream>>>(x, w, nullptr, s, bws);
  squash_kernel<<<sgrid, sblk, 0, stream>>>(s, v, 1, NS);

  caps_pass<1><<<grid, blk, 0, stream>>>(x, w, v, s, bws);
  squash_kernel<<<sgrid, sblk, 0, stream>>>(s, v, 1, NS);

  caps_pass<2><<<grid, blk, 0, stream>>>(x, w, v, s, bws);
  squash_kernel<<<sgrid, sblk, 0, stream>>>(s, out, 0, NS);
}